// WeightedSegmentationLoss_5566277616019
// MI455X (gfx1250) — compile-verified
//
#include <hip/hip_runtime.h>
#include <math.h>

// Problem constants (match reference)
static constexpr int kH = 512;
static constexpr int kW = 512;
static constexpr int kC = 6;
static constexpr float kBig = 1024.0f;          // h + w cap from reference
static constexpr float kTwoSigma2 = 200.0f;     // 2 * 10^2
static constexpr int kPixBlocks = (kH * kW) / 256;  // 1024 blocks in loss kernel

typedef float v2f __attribute__((ext_vector_type(2)));
typedef float v8f __attribute__((ext_vector_type(8)));

// ---------------------------------------------------------------------------
// Kernel 1: vertical 1-D distance scan per column (forward + backward), square.
// One thread per column; two passes over rows. Output g[h][w] = (vertical d)^2.
// ---------------------------------------------------------------------------
__global__ void edt_vertical_kernel(const float* __restrict__ target,
                                    float* __restrict__ g) {
  int j = blockIdx.x * blockDim.x + threadIdx.x;
  if (j >= kW) return;

  // forward (top -> bottom)
  float d = kBig;
  for (int i = 0; i < kH; ++i) {
    const float* t = target + ((size_t)i * kW + j) * kC;
    bool nb = (t[0] > 0.f) | (t[1] > 0.f) | (t[2] > 0.f) |
              (t[3] > 0.f) | (t[4] > 0.f);
    d = nb ? 0.0f : fminf(d + 1.0f, kBig);
    g[i * kW + j] = d;
  }
  // backward (bottom -> top), combine with forward, square
  d = kBig;
  for (int i = kH - 1; i >= 0; --i) {
    const float* t = target + ((size_t)i * kW + j) * kC;
    bool nb = (t[0] > 0.f) | (t[1] > 0.f) | (t[2] > 0.f) |
              (t[3] > 0.f) | (t[4] > 0.f);
    d = nb ? 0.0f : fminf(d + 1.0f, kBig);
    float m = fminf(g[i * kW + j], d);
    g[i * kW + j] = m * m;
  }
}

// ---------------------------------------------------------------------------
// Kernel 2: horizontal lower-envelope pass (brute force, row in LDS) + bg_w.
// One block per row; 256 threads each handle 2 x positions.
// ---------------------------------------------------------------------------
__global__ void edt_horizontal_kernel(const float* __restrict__ g,
                                      float* __restrict__ bgw) {
  __shared__ float row[kW];
  const int i = blockIdx.x;
  const int tid = threadIdx.x;

  row[tid]       = g[i * kW + tid];
  row[tid + 256] = g[i * kW + tid + 256];
  __syncthreads();

  for (int half = 0; half < 2; ++half) {
    const int x = tid + half * 256;
    float best = 3.4e38f;
#pragma unroll 8
    for (int k = 0; k < kW; ++k) {
      float dx = (float)(x - k);
      best = fminf(best, row[k] + dx * dx);
    }
    bgw[i * kW + x] = expf(-best / kTwoSigma2);
  }
}

// ---------------------------------------------------------------------------
// Kernel 3: fused soft-target CE + weighting + deterministic block reduction.
// Wave-level reduction via fp32 WMMA 16x16x4 (A = 32 lane values, B = ones):
//   D[m,n] = v_m + v_{m+16}  (replicated over n)
// Sum of a lane's 8 D VGPRs = sum of rows 0-7 (lanes 0-15) / rows 8-15
// (lanes 16-31); one shfl_xor(16) completes the 32-lane sum.
// ---------------------------------------------------------------------------
__global__ void loss_partial_kernel(const float* __restrict__ pred,
                                    const float* __restrict__ target,
                                    const float* __restrict__ bgw,
                                    float* __restrict__ partials) {
  const int idx = blockIdx.x * blockDim.x + threadIdx.x;  // pixel id
  const int HW = kH * kW;

  float p0 = pred[0 * HW + idx];
  float p1 = pred[1 * HW + idx];
  float p2 = pred[2 * HW + idx];
  float p3 = pred[3 * HW + idx];
  float p4 = pred[4 * HW + idx];
  float p5 = pred[5 * HW + idx];

  float mx = fmaxf(fmaxf(fmaxf(p0, p1), fmaxf(p2, p3)), fmaxf(p4, p5));
  float se = expf(p0 - mx) + expf(p1 - mx) + expf(p2 - mx) +
             expf(p3 - mx) + expf(p4 - mx) + expf(p5 - mx);
  float lse = mx + logf(se);

  const float* t = target + (size_t)idx * kC;
  float dot = t[0] * p0 + t[1] * p1 + t[2] * p2 +
              t[3] * p3 + t[4] * p4 + t[5] * p5;

  float ce = lse - dot;
  float val = ce * (5.0f + bgw[idx]);

  // ---- wave32 reduction through the matrix pipe (V_WMMA_F32_16X16X4_F32) ----
  v2f a;  a.x = val;  a.y = 0.0f;   // A(16x4): K=0/2 in vgpr0, K=1/3 in vgpr1
  v2f b;  b.x = 1.0f; b.y = 1.0f;   // B(4x16) = all ones
  v8f cacc = {0.f, 0.f, 0.f, 0.f, 0.f, 0.f, 0.f, 0.f};
  v8f dacc = __builtin_amdgcn_wmma_f32_16x16x4_f32(
      /*neg_a=*/false, a, /*neg_b=*/false, b,
      /*c_mod=*/(short)0, cacc, /*reuse_a=*/false, /*reuse_b=*/false);

  float s8 = dacc[0] + dacc[1] + dacc[2] + dacc[3] +
             dacc[4] + dacc[5] + dacc[6] + dacc[7];
  float wavesum = s8 + __shfl_xor(s8, 16, 32);   // combine the two row halves

  __shared__ float wsum[8];
  const int lane = threadIdx.x & 31;
  const int wv   = threadIdx.x >> 5;
  if (lane == 0) wsum[wv] = wavesum;
  __syncthreads();
  if (threadIdx.x == 0) {
    float tot = 0.0f;
#pragma unroll
    for (int k = 0; k < 8; ++k) tot += wsum[k];
    partials[blockIdx.x] = tot;
  }
}

// ---------------------------------------------------------------------------
// Kernel 4: deterministic final reduction of 1024 partials -> scalar mean.
// ---------------------------------------------------------------------------
__global__ void final_reduce_kernel(const float* __restrict__ partials,
                                    float* __restrict__ out) {
  __shared__ float sm[256];
  const int tid = threadIdx.x;
  float s = partials[tid] + partials[tid + 256] +
            partials[tid + 512] + partials[tid + 768];
  sm[tid] = s;
  __syncthreads();
  for (int off = 128; off > 0; off >>= 1) {
    if (tid < off) sm[tid] += sm[tid + off];
    __syncthreads();
  }
  if (tid == 0) out[0] = sm[0] * (1.0f / (float)(kC * kH * kW));
}

// ---------------------------------------------------------------------------
extern "C" void kernel_launch(void* const* d_in, const int* in_sizes, int n_in,
                              void* d_out, int out_size, void* d_ws, size_t ws_size,
                              hipStream_t stream) {
  const float* pred   = (const float*)d_in[0];   // [1,C,H,W] f32
  const float* target = (const float*)d_in[1];   // [1,H,W,C] f32

  float* g        = (float*)d_ws;          // H*W   (vertical d^2)
  float* bgw      = g + kH * kW;           // H*W   (background weight)
  float* partials = bgw + kH * kW;         // kPixBlocks partial sums

  edt_vertical_kernel<<<(kW + 255) / 256, 256, 0, stream>>>(target, g);
  edt_horizontal_kernel<<<kH, 256, 0, stream>>>(g, bgw);
  loss_partial_kernel<<<kPixBlocks, 256, 0, stream>>>(pred, target, bgw, partials);
  final_reduce_kernel<<<1, 256, 0, stream>>>(partials, (float*)d_out);
}